// BucklingLayer_15753940042253
// MI455X (gfx1250) — compile-verified
//
#include <hip/hip_runtime.h>
#include <math.h>

typedef __attribute__((ext_vector_type(2))) float v2f;
typedef __attribute__((ext_vector_type(8))) float v8f;

#define LMAXX 8
#define NCOEF 81
#define NPAD  84            // pad K to multiple of 4 for 16x16x4 WMMA
#define NSAMP 2000
#define PI_F  3.14159265358979323846f

// ---------------------------------------------------------------------------
// Kernel 1: spectral filter + fold SH normalization into a scaled coeff copy
// ---------------------------------------------------------------------------
__global__ void coeff_kernel(const float* __restrict__ cin,
                             const float* __restrict__ gin,
                             float* __restrict__ out,
                             float* __restrict__ wsc) {
    int k = threadIdx.x;
    if (k >= NPAD) return;
    if (k >= NCOEF) { wsc[k] = 0.f; return; }   // zero K-padding
    float g = gin[0];
    int l = 0;
    while ((l + 1) * (l + 1) <= k) l++;
    int m  = k - l * l - l;
    int am = m < 0 ? -m : m;
    // sigmoid(5*(l-2)) growth filter
    float filt = 1.f / (1.f + expf(-5.f * ((float)l - 2.f)));
    float c = cin[k] * (1.f + g * filt);
    out[k] = c;                                  // unscaled filtered coeffs
    // (l-am)!/(l+am)! as reciprocal product
    float ratio = 1.f;
    for (int i = l - am + 1; i <= l + am; ++i) ratio /= (float)i;
    float nlm   = sqrtf((2.f * l + 1.f) / (4.f * PI_F) * ratio);
    float scale = nlm * ((am > 0) ? 1.41421356237309515f : 1.f); // sqrt(2) for m!=0
    wsc[k] = c * scale;                          // scaled coeffs for the GEMVs
}

// ---------------------------------------------------------------------------
// Kernel 2: per 16-sample tile — basis (P, dP, d2P) recurrences -> LDS,
//           six f32 WMMA GEMV chains, then differential geometry.
// One wave (32 threads) per block; lanes 0-15 / 16-31 split the m-range.
// ---------------------------------------------------------------------------
__global__ void surf_kernel(const float* __restrict__ wsc,
                            float* __restrict__ out) {
    __shared__ float bas[6][16][NPAD];   // q: r, r_t, r_p, r_tt, r_tp, r_pp
    __shared__ float cbuf[NPAD];
    __shared__ float outq[6][16];

    const int lane = threadIdx.x;        // 0..31 (wave32)
    const int half = lane >> 4;
    const int row  = lane & 15;          // sample row within tile (M of WMMA)
    const int samp = blockIdx.x * 16 + row;

    // zero LDS basis (covers K padding), load scaled coeffs
    float* basf = &bas[0][0][0];
    for (int i = lane; i < 6 * 16 * NPAD; i += 32) basf[i] = 0.f;
    for (int i = lane; i < NPAD; i += 32) cbuf[i] = wsc[i];
    __syncthreads();

    // Fibonacci sphere point
    const float fidx  = (float)samp + 0.5f;
    const float phi   = (PI_F * (1.f + 2.23606797749978969f)) * fidx;
    const float ctv   = 1.f - 2.f * fidx / (float)NSAMP;
    const float theta = acosf(fminf(fmaxf(ctv, -1.f), 1.f));
    const float x = cosf(theta), s = sinf(theta);   // x=cos(th), s=sin(th)
    float sp, cp;
    sincosf(phi, &sp, &cp);

    // Each lane-half builds part of the m-range for the SAME sample row.
    const int mlo = half ? 5 : 0;
    const int mhi = half ? 8 : 4;

    // (P,Q,R) = (P_l^m, dP/dtheta, d2P/dtheta2); product-rule recurrences.
    float pmm = 1.f, qmm = 0.f, rmm = 0.f;
    float cm = 1.f, sm = 0.f;            // cos(m*phi), sin(m*phi)
    for (int m = 0; m <= LMAXX; ++m) {
        if (m > 0) {
            float cc = -(2.f * m - 1.f);
            float p0 = pmm, q0 = qmm, r0 = rmm;
            pmm = cc * s * p0;
            qmm = cc * (s * q0 + x * p0);
            rmm = cc * (s * r0 + 2.f * x * q0 - s * p0);
            float cn = cm * cp - sm * sp;
            float sn = sm * cp + cm * sp;
            cm = cn; sm = sn;
        }
        if (m >= mlo && m <= mhi) {
            float P = pmm, Q = qmm, R = rmm;
            float P2 = 0.f, Q2 = 0.f, R2 = 0.f;
            for (int l = m; l <= LMAXX; ++l) {
                if (l > m) {
                    float A1 = 2.f * l - 1.f, A2 = (float)(l + m - 1);
                    float A3i = 1.f / (float)(l - m);
                    float Pn = (A1 * (x * P) - A2 * P2) * A3i;
                    float Qn = (A1 * (x * Q - s * P) - A2 * Q2) * A3i;
                    float Rn = (A1 * (x * R - 2.f * s * Q - x * P) - A2 * R2) * A3i;
                    P2 = P; Q2 = Q; R2 = R;
                    P  = Pn; Q  = Qn; R  = Rn;
                }
                int kb = l * l + l;
                if (m == 0) {
                    bas[0][row][kb] = P;
                    bas[1][row][kb] = Q;
                    bas[3][row][kb] = R;
                } else {
                    int kc = kb + m, ks = kb - m;
                    float fm = (float)m;
                    bas[0][row][kc] = P * cm;
                    bas[1][row][kc] = Q * cm;
                    bas[2][row][kc] = -fm * P * sm;
                    bas[3][row][kc] = R * cm;
                    bas[4][row][kc] = -fm * Q * sm;
                    bas[5][row][kc] = -fm * fm * P * cm;
                    bas[0][row][ks] = P * sm;
                    bas[1][row][ks] = Q * sm;
                    bas[2][row][ks] =  fm * P * cm;
                    bas[3][row][ks] = R * sm;
                    bas[4][row][ks] =  fm * Q * cm;
                    bas[5][row][ks] = -fm * fm * P * sm;
                }
            }
        }
    }
    __syncthreads();

    // Six f32 WMMA GEMV chains: D = sum_k A(16xK) * c'[k] (B broadcast cols).
    // A layout: lane = M row, VGPR v holds K = v + 2*(lane/16).
    const int kofs = 2 * half;
    for (int q = 0; q < 6; ++q) {
        v8f acc = {0.f, 0.f, 0.f, 0.f, 0.f, 0.f, 0.f, 0.f};
#pragma unroll
        for (int kc = 0; kc < NPAD / 4; ++kc) {
            int kb = kc * 4 + kofs;
            v2f a, b;
            a.x = bas[q][row][kb];  a.y = bas[q][row][kb + 1];
            b.x = cbuf[kb];         b.y = cbuf[kb + 1];
            acc = __builtin_amdgcn_wmma_f32_16x16x4_f32(
                      false, a, false, b, (short)0, acc, false, false);
        }
        // Column 0 of D: lane 0 holds rows 0-7, lane 16 rows 8-15.
        if (lane == 0) {
#pragma unroll
            for (int j = 0; j < 8; ++j) outq[q][j] = acc[j];
        } else if (lane == 16) {
#pragma unroll
            for (int j = 0; j < 8; ++j) outq[q][8 + j] = acc[j];
        }
    }
    __syncthreads();

    // Differential geometry (lanes 0-15, one sample each)
    if (lane < 16) {
        float r   = outq[0][row];
        float rt  = outq[1][row];
        float rp  = outq[2][row];
        float rtt = outq[3][row];
        float rtp = outq[4][row];
        float rpp = outq[5][row];

        float ux = s * cp, uy = s * sp, uz = x;        // u
        float tx = x * cp, ty = x * sp, tz = -s;       // u_t
        float px = -s * sp, py = s * cp;               // u_p (z=0)
        float tpx = -x * sp, tpy = x * cp;             // u_tp (z=0)

        float Xtx = rt * ux + r * tx,  Xty = rt * uy + r * ty,  Xtz = rt * uz + r * tz;
        float Xpx = rp * ux + r * px,  Xpy = rp * uy + r * py,  Xpz = rp * uz;
        float Xttx = rtt * ux + 2.f * rt * tx - r * ux;
        float Xtty = rtt * uy + 2.f * rt * ty - r * uy;
        float Xttz = rtt * uz + 2.f * rt * tz - r * uz;
        float Xtpx = rtp * ux + rt * px + rp * tx + r * tpx;
        float Xtpy = rtp * uy + rt * py + rp * ty + r * tpy;
        float Xtpz = rtp * uz + rp * tz;
        float Xppx = rpp * ux + 2.f * rp * px - r * s * cp;
        float Xppy = rpp * uy + 2.f * rp * py - r * s * sp;
        float Xppz = rpp * uz;

        float E = Xtx * Xtx + Xty * Xty + Xtz * Xtz;
        float F = Xtx * Xpx + Xty * Xpy + Xtz * Xpz;
        float G = Xpx * Xpx + Xpy * Xpy + Xpz * Xpz;
        float cx = Xty * Xpz - Xtz * Xpy;
        float cy = Xtz * Xpx - Xtx * Xpz;
        float cz = Xtx * Xpy - Xty * Xpx;
        float inv = 1.f / (sqrtf(cx * cx + cy * cy + cz * cz) + 1e-9f);
        float nx = cx * inv, ny = cy * inv, nz = cz * inv;
        float Lf = Xttx * nx + Xtty * ny + Xttz * nz;
        float Mf = Xtpx * nx + Xtpy * ny + Xtpz * nz;
        float Nf = Xppx * nx + Xppy * ny + Xppz * nz;
        float den = E * G - F * F;
        den = den < 1e-9f ? 1e-9f : den;
        float Kg = (Lf * Nf - Mf * Mf) / den;
        float Hm = -(E * Nf + G * Lf - 2.f * F * Mf) / (2.f * den);
        float disc = fmaxf(Hm * Hm - Kg, 1e-12f);
        float si = (2.f / PI_F) * atan2f(Hm, sqrtf(disc));

        out[83 + samp]   = si;
        out[2083 + samp] = Hm;
        out[4083 + samp] = Kg;
    }
}

// ---------------------------------------------------------------------------
// Kernel 3: mean/std of si via f32 WMMA row-sum against all-ones B.
// Single wave; two accumulator chains (sum, sum of squares).
// ---------------------------------------------------------------------------
__global__ void reduce_kernel(float* __restrict__ out) {
    const float* si = out + 83;
    const int lane = threadIdx.x;
    const int half = lane >> 4;
    const int row  = lane & 15;

    v8f accS = {0.f, 0.f, 0.f, 0.f, 0.f, 0.f, 0.f, 0.f};
    v8f accQ = {0.f, 0.f, 0.f, 0.f, 0.f, 0.f, 0.f, 0.f};
    const v2f ones = {1.f, 1.f};

    for (int t = 0; t < 32; ++t) {                 // 32 * 64 = 2048 >= 2000
        int base = t * 64 + row * 4 + 2 * half;    // A[m][k] = si[t*64 + m*4 + k]
        int i0 = base     < NSAMP ? base     : NSAMP - 1;
        int i1 = base + 1 < NSAMP ? base + 1 : NSAMP - 1;
        float v0 = si[i0], v1 = si[i1];
        v0 = (base     < NSAMP) ? v0 : 0.f;        // select, no divergence at WMMA
        v1 = (base + 1 < NSAMP) ? v1 : 0.f;
        v2f a  = {v0, v1};
        v2f a2 = {v0 * v0, v1 * v1};
        accS = __builtin_amdgcn_wmma_f32_16x16x4_f32(
                   false, a,  false, ones, (short)0, accS, false, false);
        accQ = __builtin_amdgcn_wmma_f32_16x16x4_f32(
                   false, a2, false, ones, (short)0, accQ, false, false);
    }
    // every D column is identical; each lane holds 8 rows of one column
    float ls = 0.f, lq = 0.f;
#pragma unroll
    for (int j = 0; j < 8; ++j) { ls += accS[j]; lq += accQ[j]; }
    float totS = ls + __shfl_xor(ls, 16);          // rows 0-7 + rows 8-15
    float totQ = lq + __shfl_xor(lq, 16);
    if (lane == 0) {
        float mean = totS / (float)NSAMP;
        float var  = totQ / (float)NSAMP - mean * mean;
        out[81] = mean;
        out[82] = sqrtf(fmaxf(var, 0.f));
    }
}

// ---------------------------------------------------------------------------
extern "C" void kernel_launch(void* const* d_in, const int* in_sizes, int n_in,
                              void* d_out, int out_size, void* d_ws, size_t ws_size,
                              hipStream_t stream) {
    const float* cin = (const float*)d_in[0];   // coeffs_initial (81)
    const float* gin = (const float*)d_in[1];   // g (scalar)
    float* out = (float*)d_out;                 // 81+1+1+2000+2000+2000 floats
    float* wsc = (float*)d_ws;                  // 84 floats scaled coeffs

    coeff_kernel<<<1, 128, 0, stream>>>(cin, gin, out, wsc);
    surf_kernel<<<NSAMP / 16, 32, 0, stream>>>(wsc, out);
    reduce_kernel<<<1, 32, 0, stream>>>(out);
}